// CTCLayer_5669356831169
// MI455X (gfx1250) — compile-verified
//
#include <hip/hip_runtime.h>
#include <cstdint>
#include <cstddef>

// ---------------------------------------------------------------------------
// CTC batch loss, MI455X (gfx1250).
// Memory-bound: streams y_pred (512 MB) once via the Tensor Data Mover
// (tensor_load_to_lds, TENSORcnt) into triple-buffered LDS; the DP runs in a
// single wave32 per batch row with cross-lane shuffles (no barriers).
// ---------------------------------------------------------------------------

#define USE_TDM 1

#define NEGV (-1e30f)
#define EPSV (1e-7f)

constexpr int Tn   = 512;
constexpr int Cn   = 1024;
constexpr int Ln   = 64;
constexpr int CH   = 4;                 // rows per chunk (tile_dim1)
constexpr int NBUF = 3;                 // LDS row buffers
constexpr int NCHUNK = Tn / CH;         // 128
constexpr int LOADS_PER_CHUNK = (CH * Cn * 4) / (32 * 16);   // 32 (async path)

typedef unsigned int v4u __attribute__((ext_vector_type(4)));
typedef unsigned int v8u __attribute__((ext_vector_type(8)));

#if USE_TDM
// One TDM descriptor moves a whole CH x Cn f32 tile (16 KB) global -> LDS.
__device__ __forceinline__ void tdm_load_chunk(const void* gptr, unsigned lds_addr) {
  const unsigned long long ga = (unsigned long long)(uintptr_t)gptr;
  v4u g0;
  g0.x = 1u;                                        // count=1, user mode, no gather
  g0.y = lds_addr;                                  // lds_addr (bytes)
  g0.z = (unsigned)(ga & 0xFFFFFFFFu);              // global_addr[31:0]
  g0.w = (unsigned)((ga >> 32) & 0x01FFFFFFu)       // global_addr[56:32]
       | (2u << 30);                                // type = 2 ("image")
  v8u g1;
  g1.s0 = (2u << 16);                               // data_size=2 (4 B); no multicast/pad
  g1.s1 = ((unsigned)Cn & 0xFFFFu) << 16;           // tensor_dim0[15:0] @ bits 63:48
  g1.s2 = (((unsigned)Cn >> 16) & 0xFFFFu)          // tensor_dim0[31:16]
        | (((unsigned)Tn & 0xFFFFu) << 16);         // tensor_dim1[15:0]
  g1.s3 = (((unsigned)Tn >> 16) & 0xFFFFu)          // tensor_dim1[31:16]
        | (((unsigned)Cn & 0xFFFFu) << 16);         // tile_dim0 = 1024
  g1.s4 = (unsigned)CH;                             // tile_dim1 = 4, tile_dim2 = 0
  g1.s5 = (unsigned)Cn;                             // tensor_dim0_stride[31:0] = 1024
  g1.s6 = 0u;                                       // stride hi / dim1_stride lo
  g1.s7 = 0u;
  // 2-group form (2-D tensor): VADDR2/VADDR3 = NULL
  asm volatile("tensor_load_to_lds %0, %1" :: "s"(g0), "s"(g1) : "memory");
}

template <int N>
__device__ __forceinline__ void wait_stream() {
#if __has_builtin(__builtin_amdgcn_s_wait_tensorcnt)
  __builtin_amdgcn_s_wait_tensorcnt((short)N);
#else
  asm volatile("s_wait_tensorcnt %0" :: "n"(N) : "memory");
#endif
}
#else
__device__ __forceinline__ void async_copy_b128(unsigned lds_off, const void* gptr) {
  asm volatile("global_load_async_to_lds_b128 %0, %1, off"
               :: "v"(lds_off), "v"(gptr) : "memory");
}
template <int N>
__device__ __forceinline__ void wait_stream() {
#if __has_builtin(__builtin_amdgcn_s_wait_asynccnt)
  __builtin_amdgcn_s_wait_asynccnt((short)N);
#else
  asm volatile("s_wait_asynccnt %0" :: "n"(N) : "memory");
#endif
}
#endif

__device__ __forceinline__ void issue_chunk(const float* gbase, unsigned lds_byte_base, int lane) {
#if USE_TDM
  (void)lane;
  tdm_load_chunk((const void*)gbase, lds_byte_base);
#else
#pragma unroll
  for (int j = 0; j < LOADS_PER_CHUNK; ++j) {
    const unsigned off = (unsigned)((lane + j * 32) * 16);
    async_copy_b128(lds_byte_base + off, (const char*)gbase + off);
  }
#endif
}

__device__ __forceinline__ void wait_dscnt0() {
#if __has_builtin(__builtin_amdgcn_s_wait_dscnt)
  __builtin_amdgcn_s_wait_dscnt(0);
#else
  asm volatile("s_wait_dscnt 0" ::: "memory");
#endif
}

// per chunk in flight: TDM = 1 descriptor, async path = 32 b128 ops
#if USE_TDM
constexpr int INFLIGHT_PER_CHUNK = 1;
#else
constexpr int INFLIGHT_PER_CHUNK = LOADS_PER_CHUNK;
#endif

__global__ __launch_bounds__(32)
void ctc_loss_kernel(const int* __restrict__ y_true,
                     const float* __restrict__ y_pred,
                     float* __restrict__ out) {
  __shared__ float rows[NBUF * CH * Cn];   // 48 KB row stream buffers
  __shared__ float afin[160];              // final alpha spill

  const int b     = blockIdx.x;
  const int lane  = threadIdx.x & 31;
  const int blank = Cn - 1;

  const float* yp_b = y_pred + (size_t)b * Tn * Cn;
  const int*   yt   = y_true + (size_t)b * Ln;

  const unsigned rows_base = (unsigned)(uintptr_t)(const void*)&rows[0];

  // Kick off the stream: chunks 0 and 1 in flight before any compute.
  issue_chunk(yp_b + 0 * CH * Cn, rows_base + (unsigned)(0 * CH * Cn * 4), lane);
  issue_chunk(yp_b + 1 * CH * Cn, rows_base + (unsigned)((1 % NBUF) * CH * Cn * 4), lane);

  // ---- per-row label metadata (all lanes read the same 64 ints: broadcast) --
  int ll = 0;
  for (int i = 0; i < Ln; ++i) ll += (yt[i] != -1) ? 1 : 0;
  const int Smax = 2 * ll + 1;

  int  cls[5];
  bool allow2k[5];
  bool validk[5];
#pragma unroll
  for (int k = 0; k < 5; ++k) {
    const int s = 5 * lane + k;
    validk[k] = (s < Smax) && (s <= 2 * Ln);
    int  c  = blank;
    bool a2 = false;
    if (s & 1) {
      const int li = s >> 1;
      if (li < Ln) {
        int lab = yt[li];
        if (lab < 0) lab = blank;
        c = lab;
        if (li >= 1) {
          int lab2 = yt[li - 1];
          if (lab2 < 0) lab2 = blank;
          a2 = (lab != lab2) && (lab != blank);
        }
      }
    }
    cls[k]     = validk[k] ? c : blank;
    allow2k[k] = a2 && (s >= 2);
  }

  // ---- DP over time, consuming LDS rows as the TDM stream lands -----------
  float a[5];
#pragma unroll
  for (int k = 0; k < 5; ++k) a[k] = NEGV;

  for (int ck = 0; ck < NCHUNK; ++ck) {
    // chunk ck complete when at most chunk ck+1's transfer remains outstanding
    wait_stream<INFLIGHT_PER_CHUNK>();
    if (ck + 2 < NCHUNK) {
      issue_chunk(yp_b + (size_t)(ck + 2) * CH * Cn,
                  rows_base + (unsigned)(((ck + 2) % NBUF) * CH * Cn * 4), lane);
    }
    const float* rowbase = &rows[(ck % NBUF) * CH * Cn];

    for (int r = 0; r < CH; ++r) {
      const int t = ck * CH + r;
      const float* row = rowbase + r * Cn;

      float prevA4 = __shfl_up(a[4], 1, 32);
      float prevA3 = __shfl_up(a[3], 1, 32);
      if (lane == 0) { prevA4 = NEGV; prevA3 = NEGV; }

      if (t == 0) {
#pragma unroll
        for (int k = 0; k < 5; ++k) {
          const int s = 5 * lane + k;
          const float e = __logf(row[cls[k]] + EPSV);
          a[k] = ((s < 2) && validk[k]) ? e : NEGV;
        }
      } else {
        float na[5];
#pragma unroll
        for (int k = 0; k < 5; ++k) {
          const float am1 = (k == 0) ? prevA4 : a[k - 1];
          float am2 = (k == 0) ? prevA3 : ((k == 1) ? prevA4 : a[k - 2]);
          am2 = allow2k[k] ? am2 : NEGV;
          const float m  = fmaxf(fmaxf(a[k], am1), am2);
          const float l3 = m + __logf(__expf(a[k] - m) + __expf(am1 - m) + __expf(am2 - m));
          const float e  = __logf(row[cls[k]] + EPSV);
          na[k] = validk[k] ? (e + l3) : NEGV;
        }
#pragma unroll
        for (int k = 0; k < 5; ++k) a[k] = na[k];
      }
    }
  }

  // ---- finalize: -logaddexp(alpha[2ll], alpha[2ll-1]) ---------------------
#pragma unroll
  for (int k = 0; k < 5; ++k) {
    const int s = 5 * lane + k;
    if (s < 160) afin[s] = a[k];
  }
  wait_dscnt0();   // single wave: DS in-order; make cross-lane spill visible

  if (lane == 0) {
    const float aL = afin[2 * ll];
    const float aP = (ll > 0) ? afin[2 * ll - 1] : NEGV;
    const float m  = fmaxf(aL, aP);
    out[b] = -(m + __logf(__expf(aL - m) + __expf(aP - m)));
  }
}

extern "C" void kernel_launch(void* const* d_in, const int* in_sizes, int n_in,
                              void* d_out, int out_size, void* d_ws, size_t ws_size,
                              hipStream_t stream) {
  (void)n_in; (void)d_ws; (void)ws_size; (void)out_size;
  const int*   y_true = (const int*)d_in[0];    // [B, 64] int32
  const float* y_pred = (const float*)d_in[1];  // [B, 512, 1024] float32
  float*       out    = (float*)d_out;          // [B, 1] float32

  const int B = in_sizes[0] / Ln;               // 256
  ctc_loss_kernel<<<B, 32, 0, stream>>>(y_true, y_pred, out);
}